// Block_59760174956851
// MI455X (gfx1250) — compile-verified
//
#include <hip/hip_runtime.h>

typedef __attribute__((ext_vector_type(2))) float v2f;
typedef __attribute__((ext_vector_type(8))) float v8f;

#define NVOX 400000
#define NIN  64
#define NOUT 64
#define KOFF 27
#define LDSP 68              // padded LDS column stride (dwords) -> conflict-free b64 reads
#define LDSBUF (NOUT * LDSP) // one weight tile buffer (4352 dwords)
#define LEAK (1.0f / 3.0f)
#define EPS  1e-4f

// ---------------------------------------------------------------------------
// Kernel 1: rulebook sparse conv via fp32 WMMA, bias add, per-channel partial
// BN stats (sum, sumsq) accumulated atomically into red[0..127].
// Block = 128 threads = 4 waves; wave w handles rows [blk*64 + w*16, +16).
// Weight tiles are register-pipelined into double-buffered LDS: one barrier
// per offset, W[k+2] global loads overlap compute of offset k.
// ---------------------------------------------------------------------------
__global__ __launch_bounds__(128)
void spconv_wmma_kernel(const float* __restrict__ feat,
                        const float* __restrict__ Wg,
                        const float* __restrict__ bias,
                        const int*   __restrict__ nbr,
                        float*       __restrict__ out,
                        float*       __restrict__ red)
{
    __shared__ float lds[2 * LDSBUF];        // W tile double buffer, transposed:
                                             // buf[nout*LDSP + kin]
    const int tid  = threadIdx.x;
    const int wave = tid >> 5;
    const int lane = tid & 31;
    const int li   = lane & 15;              // index within half-wave
    const int half = lane >> 4;              // 0 or 1
    const int mbase = blockIdx.x * 64 + wave * 16;
    const int row   = mbase + li;

    const float4* wk4 = reinterpret_cast<const float4*>(Wg); // 1024 float4 per offset

    float4 wreg[8];
    auto wload = [&](int k) {                // W[k] -> registers (coalesced b128)
        #pragma unroll
        for (int it = 0; it < 8; ++it)
            wreg[it] = wk4[k * 1024 + it * 128 + tid];
    };
    auto wstage = [&](float* dst) {          // registers -> LDS, transposed
        #pragma unroll
        for (int it = 0; it < 8; ++it) {
            int e   = (it * 128 + tid) * 4;
            int kin = e >> 6;
            int n0  = e & 63;                // multiple of 4, same kin for all 4
            float4 v = wreg[it];
            dst[(n0 + 0) * LDSP + kin] = v.x;
            dst[(n0 + 1) * LDSP + kin] = v.y;
            dst[(n0 + 2) * LDSP + kin] = v.z;
            dst[(n0 + 3) * LDSP + kin] = v.w;
        }
    };

    v8f acc[4];
    #pragma unroll
    for (int t = 0; t < 4; ++t)
        acc[t] = (v8f){0.f, 0.f, 0.f, 0.f, 0.f, 0.f, 0.f, 0.f};

    // ---- pipeline prologue: buf0 <- W[0]; regs <- W[1]; idx for k=0 ----
    wload(0);
    wstage(lds);
    wload(1);
    int idx = nbr[row * KOFF + 0];

    for (int k = 0; k < KOFF; ++k) {
        __syncthreads();                     // all waves done with buf[(k+1)&1]
                                             // and buf[k&1] stores are visible
        if (k + 1 < KOFF)
            wstage(lds + ((k + 1) & 1) * LDSBUF);
        if (k + 2 < KOFF)
            wload(k + 2);                    // latency hidden under compute

        int idx_next = NVOX;
        if (k + 1 < KOFF) {
            idx_next = nbr[row * KOFF + k + 1];
            const float* pf = feat + (size_t)(idx_next < NVOX ? idx_next : 0) * NIN;
            __builtin_prefetch(pf, 0, 0);            // global_prefetch_b8
            __builtin_prefetch(pf + 32, 0, 0);       // second 128B line of row
        }

        const float* buf   = lds + (k & 1) * LDSBUF;
        bool         valid = (idx < NVOX);   // idx == NVOX means missing neighbor
        const float* arow  = feat + (size_t)(valid ? idx : 0) * NIN + half * 2;

        #pragma unroll
        for (int c = 0; c < NIN; c += 4) {
            // A fragment (16x4 f32): lanes 0-15 -> K=c,c+1 ; lanes 16-31 -> K=c+2,c+3
            v2f a = *reinterpret_cast<const v2f*>(arow + c);
            a.x = valid ? a.x : 0.f;
            a.y = valid ? a.y : 0.f;
            #pragma unroll
            for (int t = 0; t < 4; ++t) {
                // B fragment (4x16 f32): lane -> col = t*16+li, K pair = c+2*half
                v2f b = *reinterpret_cast<const v2f*>(&buf[(t * 16 + li) * LDSP + c + half * 2]);
                acc[t] = __builtin_amdgcn_wmma_f32_16x16x4_f32(
                    /*neg_a=*/false, a, /*neg_b=*/false, b,
                    /*c_mod=*/(short)0, acc[t],
                    /*reuse_a=*/false, /*reuse_b=*/false);
            }
        }
        idx = idx_next;
    }

    // ---- epilogue: bias, store pre-BN output, per-channel partial stats ----
    float psum[4], psq[4];
    #pragma unroll
    for (int t = 0; t < 4; ++t) {
        int   col = t * 16 + li;
        float b   = bias[col];
        float s = 0.f, q = 0.f;
        #pragma unroll
        for (int j = 0; j < 8; ++j) {        // C layout: VGPR j -> M = j + 8*half
            float v = acc[t][j] + b;
            out[(size_t)(mbase + half * 8 + j) * NOUT + col] = v;
            s += v;
            q += v * v;
        }
        psum[t] = s;
        psq[t]  = q;
    }

    __syncthreads();                          // done reading W tiles from LDS
    #pragma unroll
    for (int t = 0; t < 4; ++t) {             // fold the two half-waves (same col)
        psum[t] += __shfl_xor(psum[t], 16, 32);
        psq[t]  += __shfl_xor(psq[t],  16, 32);
    }
    float* rs = lds;                          // [4][64] sums
    float* rq = lds + 4 * 64;                 // [4][64] sumsqs
    if (half == 0) {
        #pragma unroll
        for (int t = 0; t < 4; ++t) {
            rs[wave * 64 + t * 16 + li] = psum[t];
            rq[wave * 64 + t * 16 + li] = psq[t];
        }
    }
    __syncthreads();
    if (tid < 64) {
        float s = rs[tid] + rs[64 + tid] + rs[128 + tid] + rs[192 + tid];
        float q = rq[tid] + rq[64 + tid] + rq[128 + tid] + rq[192 + tid];
        atomicAdd(&red[tid], s);
        atomicAdd(&red[64 + tid], q);
    }
}

// ---------------------------------------------------------------------------
// Kernel 2: fold BN stats + gamma/beta into per-channel scale/shift.
// ---------------------------------------------------------------------------
__global__ void bn_stats_kernel(const float* __restrict__ red,
                                const float* __restrict__ gamma,
                                const float* __restrict__ beta,
                                float* __restrict__ coef)
{
    int c = threadIdx.x;
    if (c < NOUT) {
        float mean = red[c] * (1.f / NVOX);
        float var  = red[64 + c] * (1.f / NVOX) - mean * mean;
        float inv  = rsqrtf(var + EPS);
        float sc   = gamma[c] * inv;
        coef[c]      = sc;
        coef[64 + c] = beta[c] - mean * sc;
    }
}

// ---------------------------------------------------------------------------
// Kernel 3: in-place BN apply + LeakyReLU, float4 vectorized.
// ---------------------------------------------------------------------------
__global__ __launch_bounds__(256)
void bn_act_kernel(float* __restrict__ out, const float* __restrict__ coef)
{
    int i  = blockIdx.x * 256 + threadIdx.x;  // float4 index; total = NVOX*NOUT/4
    int c4 = (i * 4) & 63;                    // multiple of 4 -> aligned coef loads
    float4 x = reinterpret_cast<float4*>(out)[i];
    float4 sc = *reinterpret_cast<const float4*>(coef + c4);
    float4 sh = *reinterpret_cast<const float4*>(coef + 64 + c4);
    float4 y;
    y.x = sc.x * x.x + sh.x;
    y.y = sc.y * x.y + sh.y;
    y.z = sc.z * x.z + sh.z;
    y.w = sc.w * x.w + sh.w;
    y.x = y.x > 0.f ? y.x : y.x * LEAK;
    y.y = y.y > 0.f ? y.y : y.y * LEAK;
    y.z = y.z > 0.f ? y.z : y.z * LEAK;
    y.w = y.w > 0.f ? y.w : y.w * LEAK;
    reinterpret_cast<float4*>(out)[i] = y;
}

// ---------------------------------------------------------------------------
extern "C" void kernel_launch(void* const* d_in, const int* in_sizes, int n_in,
                              void* d_out, int out_size, void* d_ws, size_t ws_size,
                              hipStream_t stream)
{
    const float* feat  = (const float*)d_in[0];   // [N,64]
    const float* Wg    = (const float*)d_in[1];   // [27,64,64]
    const float* bias  = (const float*)d_in[2];   // [64]
    const float* gamma = (const float*)d_in[3];   // [64]
    const float* beta  = (const float*)d_in[4];   // [64]
    const int*   nbr   = (const int*)d_in[5];     // [N,27]
    float* out = (float*)d_out;                   // [N,64]
    float* wsf = (float*)d_ws;                    // [0..127]=stats, [128..255]=coef

    hipMemsetAsync(wsf, 0, 128 * sizeof(float), stream);

    spconv_wmma_kernel<<<NVOX / 64, 128, 0, stream>>>(feat, Wg, bias, nbr, out, wsf);
    bn_stats_kernel<<<1, 64, 0, stream>>>(wsf, gamma, beta, wsf + 128);
    bn_act_kernel<<<(NVOX * NOUT / 4) / 256, 256, 0, stream>>>(out, wsf + 128);
}